// RelGraphAttentionHetero_25890062860618
// MI455X (gfx1250) — compile-verified
//
#include <hip/hip_runtime.h>
#include <math.h>

// ---------------------------------------------------------------------------
// Types for WMMA fragments (wave32, gfx1250)
// ---------------------------------------------------------------------------
typedef __attribute__((ext_vector_type(2))) float v2f;
typedef __attribute__((ext_vector_type(8))) float v8f;

// ---------------------------------------------------------------------------
// Helpers
// ---------------------------------------------------------------------------
__device__ inline void atomicMaxF32(float* addr, float val) {
    // Int-punned float max: works with -inf init.
    int vi = __float_as_int(val);
    if (vi >= 0) {
        atomicMax((int*)addr, vi);
    } else {
        atomicMin((unsigned int*)addr, __float_as_uint(val));
    }
}

// ---------------------------------------------------------------------------
// Kernel 0: init agg (in d_out) = 0, m = -inf, denom = 0
// ---------------------------------------------------------------------------
__global__ __launch_bounds__(256) void init_buffers(float* __restrict__ out,
                                                    float* __restrict__ m,
                                                    float* __restrict__ denom,
                                                    long long total_out, int rn) {
    long long idx = (long long)blockIdx.x * blockDim.x + threadIdx.x;
    if (idx < total_out) out[idx] = 0.0f;
    if (idx < rn) {
        m[idx] = -INFINITY;
        denom[idx] = 0.0f;
    }
}

// ---------------------------------------------------------------------------
// Kernel 1: per-edge logits. One wave32 per edge; lane covers 4 floats.
//   e[r,i] = dot(x[src], a[:D]) + dot(x[dst], a[D:])
// ---------------------------------------------------------------------------
__global__ __launch_bounds__(256) void edge_logits(const float* __restrict__ x,
                                                   const float* __restrict__ attn_w,
                                                   const int* __restrict__ src,
                                                   const int* __restrict__ dst,
                                                   float* __restrict__ e,
                                                   int E, int D) {
    const int lane = threadIdx.x & 31;
    const int edge = blockIdx.x * 8 + (threadIdx.x >> 5);
    const int r = blockIdx.y;
    if (edge >= E) return;

    const int s = src[(size_t)r * E + edge];
    const int t = dst[(size_t)r * E + edge];
    const float* a0 = attn_w + (size_t)r * 2 * D;
    const float* a1 = a0 + D;
    const float* xs = x + (size_t)s * D;
    const float* xd = x + (size_t)t * D;

    float acc = 0.0f;
    for (int d = lane * 4; d < D; d += 128) {
        float4 vs = *(const float4*)(xs + d);
        float4 vd = *(const float4*)(xd + d);
        float4 w0 = *(const float4*)(a0 + d);
        float4 w1 = *(const float4*)(a1 + d);
        acc += vs.x * w0.x + vs.y * w0.y + vs.z * w0.z + vs.w * w0.w;
        acc += vd.x * w1.x + vd.y * w1.y + vd.z * w1.z + vd.w * w1.w;
    }
    // wave32 tree reduction
    for (int off = 16; off > 0; off >>= 1) acc += __shfl_xor(acc, off, 32);
    if (lane == 0) e[(size_t)r * E + edge] = acc;
}

// ---------------------------------------------------------------------------
// Kernel 2: per-destination segment max (per relation)
// ---------------------------------------------------------------------------
__global__ __launch_bounds__(256) void seg_max(const float* __restrict__ e,
                                               const int* __restrict__ dst,
                                               float* __restrict__ m,
                                               int E, int N) {
    const int idx = blockIdx.x * blockDim.x + threadIdx.x;
    const int r = blockIdx.y;
    if (idx >= E) return;
    const int t = dst[(size_t)r * E + idx];
    atomicMaxF32(&m[(size_t)r * N + t], e[(size_t)r * E + idx]);
}

// ---------------------------------------------------------------------------
// Kernel 3: p = exp(e - m[dst]); denom[dst] += p  (p overwrites e)
// ---------------------------------------------------------------------------
__global__ __launch_bounds__(256) void exp_sum(float* __restrict__ e,
                                               const int* __restrict__ dst,
                                               const float* __restrict__ m,
                                               float* __restrict__ denom,
                                               int E, int N) {
    const int idx = blockIdx.x * blockDim.x + threadIdx.x;
    const int r = blockIdx.y;
    if (idx >= E) return;
    const int t = dst[(size_t)r * E + idx];
    const float p = __expf(e[(size_t)r * E + idx] - m[(size_t)r * N + t]);
    e[(size_t)r * E + idx] = p;
    atomicAdd(&denom[(size_t)r * N + t], p);
}

// ---------------------------------------------------------------------------
// Kernel 4: weighted scatter: agg[dst] += (p / denom[dst]) * x[src]
//   One wave32 per edge, 4 floats per lane.
// ---------------------------------------------------------------------------
__global__ __launch_bounds__(256) void scatter_agg(const float* __restrict__ x,
                                                   const float* __restrict__ p,
                                                   const float* __restrict__ denom,
                                                   const int* __restrict__ src,
                                                   const int* __restrict__ dst,
                                                   float* __restrict__ out,
                                                   int E, int D, int N) {
    const int lane = threadIdx.x & 31;
    const int edge = blockIdx.x * 8 + (threadIdx.x >> 5);
    const int r = blockIdx.y;
    if (edge >= E) return;

    const int s = src[(size_t)r * E + edge];
    const int t = dst[(size_t)r * E + edge];
    const float alpha = p[(size_t)r * E + edge] /
                        fmaxf(denom[(size_t)r * N + t], 1e-30f);
    const float* xs = x + (size_t)s * D;
    float* od = out + (size_t)t * D;
    for (int d = lane * 4; d < D; d += 128) {
        float4 vs = *(const float4*)(xs + d);
        atomicAdd(od + d + 0, alpha * vs.x);
        atomicAdd(od + d + 1, alpha * vs.y);
        atomicAdd(od + d + 2, alpha * vs.z);
        atomicAdd(od + d + 3, alpha * vs.w);
    }
}

// ---------------------------------------------------------------------------
// Kernel 5: self-loop GEMM via V_WMMA_F32_16X16X4_F32 (full f32 precision).
//   One wave computes a 16x64 output slab: 4 C accumulators share one A frag.
//   out[m,n] = agg(out) + sum_k x[m,k]*W[k,n] + bias[n]
// Fragment layouts per CDNA5 ISA 7.12.2:
//   A (16x4 f32):  lane holds M=lane&15, VGPR j holds K=(lane>>4)*2+j
//   B (4x16 f32):  lane holds N=lane&15, VGPR j holds K=(lane>>4)*2+j
//   C (16x16 f32): VGPR v holds M = v + 8*(lane>>4), N = lane&15
// ---------------------------------------------------------------------------
__global__ __launch_bounds__(32) void wmma_selfloop(const float* __restrict__ x,
                                                    const float* __restrict__ W,
                                                    const float* __restrict__ bias,
                                                    float* __restrict__ out,
                                                    int N, int D) {
    const int lane = threadIdx.x;           // 0..31, all lanes active (EXEC all 1s)
    const int m0 = blockIdx.x * 16;
    const int n0 = blockIdx.y * 64;
    const int row = lane & 15;
    const int khalf = (lane >> 4) * 2;      // 0 or 2

    v8f c0 = {}, c1 = {}, c2 = {}, c3 = {};

    for (int k = 0; k < D; k += 4) {
        // A fragment: two consecutive K values -> one contiguous float2 load
        const float* ap = x + (size_t)(m0 + row) * D + (k + khalf);
        v2f a;
        a.x = ap[0];
        a.y = ap[1];

        // B fragments for 4 consecutive 16-wide column tiles
        const float* bp = W + (size_t)(k + khalf) * D + n0 + row;
        v2f b0, b1, b2, b3;
        b0.x = bp[0];  b0.y = bp[D + 0];
        b1.x = bp[16]; b1.y = bp[D + 16];
        b2.x = bp[32]; b2.y = bp[D + 32];
        b3.x = bp[48]; b3.y = bp[D + 48];

        c0 = __builtin_amdgcn_wmma_f32_16x16x4_f32(false, a, false, b0, (short)0, c0, false, false);
        c1 = __builtin_amdgcn_wmma_f32_16x16x4_f32(false, a, false, b1, (short)0, c1, false, false);
        c2 = __builtin_amdgcn_wmma_f32_16x16x4_f32(false, a, false, b2, (short)0, c2, false, false);
        c3 = __builtin_amdgcn_wmma_f32_16x16x4_f32(false, a, false, b3, (short)0, c3, false, false);
    }

    // Epilogue: out (currently holds agg) += gemm + bias
    const int col = n0 + (lane & 15);
    const int mbase = m0 + (lane >> 4) * 8;
#pragma unroll
    for (int v = 0; v < 8; ++v) {
        float* o = out + (size_t)(mbase + v) * D;
        o[col + 0]  += c0[v] + bias[col + 0];
        o[col + 16] += c1[v] + bias[col + 16];
        o[col + 32] += c2[v] + bias[col + 32];
        o[col + 48] += c3[v] + bias[col + 48];
    }
}

// ---------------------------------------------------------------------------
// Scalar tail for rows not covered by 16-row tiles (none for N=100000)
// ---------------------------------------------------------------------------
__global__ __launch_bounds__(256) void gemm_tail(const float* __restrict__ x,
                                                 const float* __restrict__ W,
                                                 const float* __restrict__ bias,
                                                 float* __restrict__ out,
                                                 int N, int D, int row0) {
    long long idx = (long long)blockIdx.x * blockDim.x + threadIdx.x;
    int row = row0 + (int)(idx / D);
    int col = (int)(idx % D);
    if (row >= N) return;
    float acc = out[(size_t)row * D + col] + bias[col];
    for (int k = 0; k < D; ++k) acc += x[(size_t)row * D + k] * W[(size_t)k * D + col];
    out[(size_t)row * D + col] = acc;
}

// ---------------------------------------------------------------------------
// Launcher
// ---------------------------------------------------------------------------
extern "C" void kernel_launch(void* const* d_in, const int* in_sizes, int n_in,
                              void* d_out, int out_size, void* d_ws, size_t ws_size,
                              hipStream_t stream) {
    const float* x      = (const float*)d_in[0];
    const float* attn_w = (const float*)d_in[1];
    const float* W      = (const float*)d_in[2];
    const float* bias   = (const float*)d_in[3];
    const int*   src    = (const int*)d_in[4];
    const int*   dst    = (const int*)d_in[5];
    float*       out    = (float*)d_out;

    const int D = in_sizes[3];                 // 128
    const int N = in_sizes[0] / D;             // 100000
    const int R = in_sizes[1] / (2 * D);       // 2
    const int E = in_sizes[4] / R;             // 300000

    // Workspace carve-out: e/p [R*E], m [R*N], denom [R*N]
    float* e_buf   = (float*)d_ws;
    float* m_buf   = e_buf + (size_t)R * E;
    float* den_buf = m_buf + (size_t)R * N;

    // 0) init
    {
        long long total_out = (long long)N * D;
        int rn = R * N;
        long long mx = total_out > rn ? total_out : rn;
        int blocks = (int)((mx + 255) / 256);
        init_buffers<<<dim3(blocks), dim3(256), 0, stream>>>(out, m_buf, den_buf,
                                                             total_out, rn);
    }

    dim3 wgrid((E + 7) / 8, R);     // wave-per-edge kernels
    dim3 tgrid((E + 255) / 256, R); // thread-per-edge kernels

    // 1) edge logits
    edge_logits<<<wgrid, 256, 0, stream>>>(x, attn_w, src, dst, e_buf, E, D);
    // 2) segment max
    seg_max<<<tgrid, 256, 0, stream>>>(e_buf, dst, m_buf, E, N);
    // 3) exp + segment sum
    exp_sum<<<tgrid, 256, 0, stream>>>(e_buf, dst, m_buf, den_buf, E, N);
    // 4) weighted scatter into out (= agg)
    scatter_agg<<<wgrid, 256, 0, stream>>>(x, e_buf, den_buf, src, dst, out, E, D, N);

    // 5) self-loop GEMM + bias epilogue (WMMA path when tiles divide evenly)
    int row_tiles = 0;
    if ((D % 64) == 0) {
        row_tiles = N / 16;
        if (row_tiles > 0) {
            dim3 ggrid(row_tiles, D / 64);
            wmma_selfloop<<<ggrid, 32, 0, stream>>>(x, W, bias, out, N, D);
        }
    }
    int row0 = row_tiles * 16;
    long long tail_elems = (long long)(N - row0) * D;
    if (tail_elems > 0) {
        int blocks = (int)((tail_elems + 255) / 256);
        gemm_tail<<<dim3(blocks), dim3(256), 0, stream>>>(x, W, bias, out, N, D, row0);
    }
}